// Octree2Col_29265907155618
// MI455X (gfx1250) — compile-verified
//
#include <hip/hip_runtime.h>

// Octree2Col gather: out[c][k][n] = (neigh[n][k] >= 0) ? data_in[c][neigh[n][k]] : 0
// C=32 channels, K=27 neighbors, N spatial. Pure gather -> store-bandwidth bound.
// data_in (32 MB) stays resident in the 192 MB L2; output (906 MB) streamed with
// non-temporal stores so it does not evict the gather table.

#define OCT_C   32
#define OCT_K   27
#define VEC     4
#define BLOCK   256
#define TILE_N  (BLOCK * VEC)   // 1024 n per block

typedef __attribute__((ext_vector_type(4))) float v4f;

#if defined(__HIP_DEVICE_COMPILE__) && defined(__gfx1250__) && \
    __has_builtin(__builtin_amdgcn_global_load_async_to_lds_b32)
#define HAVE_ASYNC_LDS 1
#else
#define HAVE_ASYNC_LDS 0
#endif

__global__ __launch_bounds__(BLOCK) void octree2col_gather(
    const float* __restrict__ data_in,   // (C, N)
    const int*   __restrict__ neigh,     // (N, K) int32
    float*       __restrict__ out,       // (C, K, N)
    int N)
{
  const int k   = blockIdx.y;
  const int tid = threadIdx.x;
  const int n0  = blockIdx.x * TILE_N + tid * VEC;
  if (n0 + (VEC - 1) >= N) return;      // N is a multiple of TILE_N here

  // ---- fetch the 4 neighbor indices for (n0..n0+3, k) -------------------
  int nv0, nv1, nv2, nv3;
#if HAVE_ASYNC_LDS
  __shared__ int s_nb[TILE_N];
  {
    const int* g = neigh + (size_t)n0 * OCT_K + k;   // stride K*4 B per n
    int*       l = &s_nb[tid * VEC];
    // Builtin signature (from hipcc diagnostic): param0 is global-AS int*,
    // non-const. Cast away const + address-space in one C-style cast.
    typedef __attribute__((address_space(1))) int* gip;
    typedef __attribute__((address_space(3))) int* lip;
    // gfx1250 async DMA: global -> LDS, tracked with ASYNCcnt
    __builtin_amdgcn_global_load_async_to_lds_b32((gip)(g            ), (lip)(l    ), 0, 0);
    __builtin_amdgcn_global_load_async_to_lds_b32((gip)(g +     OCT_K), (lip)(l + 1), 0, 0);
    __builtin_amdgcn_global_load_async_to_lds_b32((gip)(g + 2 * OCT_K), (lip)(l + 2), 0, 0);
    __builtin_amdgcn_global_load_async_to_lds_b32((gip)(g + 3 * OCT_K), (lip)(l + 3), 0, 0);
#if __has_builtin(__builtin_amdgcn_s_wait_asynccnt)
    __builtin_amdgcn_s_wait_asynccnt(0);
#else
    asm volatile("s_wait_asynccnt 0" ::: "memory");
#endif
    // each thread only reads back what its own wave's async ops wrote
    nv0 = l[0]; nv1 = l[1]; nv2 = l[2]; nv3 = l[3];
  }
#else
  {
    const int* g = neigh + (size_t)n0 * OCT_K + k;
    nv0 = __builtin_nontemporal_load(g);             // read-once: NT hint
    nv1 = __builtin_nontemporal_load(g +     OCT_K);
    nv2 = __builtin_nontemporal_load(g + 2 * OCT_K);
    nv3 = __builtin_nontemporal_load(g + 3 * OCT_K);
  }
#endif

  const bool b0 = nv0 >= 0, b1 = nv1 >= 0, b2 = nv2 >= 0, b3 = nv3 >= 0;
  const int  i0 = b0 ? nv0 : 0;
  const int  i1 = b1 ? nv1 : 0;
  const int  i2 = b2 ? nv2 : 0;
  const int  i3 = b3 ? nv3 : 0;

  // ---- gather 32 channels, select-mask, stream out with NT stores -------
  float* obase = out + (size_t)k * N + n0;           // then step c by K*N
#pragma unroll
  for (int c = 0; c < OCT_C; ++c) {
    const float* __restrict__ row = data_in + (size_t)c * N;
    // unconditional L2-resident gathers (idx always in-bounds), then cndmask
    float t0 = row[i0];
    float t1 = row[i1];
    float t2 = row[i2];
    float t3 = row[i3];
    v4f v;
    v.x = b0 ? t0 : 0.0f;
    v.y = b1 ? t1 : 0.0f;
    v.z = b2 ? t2 : 0.0f;
    v.w = b3 ? t3 : 0.0f;
    // 16B-aligned (n0 % 4 == 0, N % 4 == 0), wave-contiguous, non-temporal
    __builtin_nontemporal_store(v, (v4f*)(obase + (size_t)c * OCT_K * N));
  }
}

extern "C" void kernel_launch(void* const* d_in, const int* in_sizes, int n_in,
                              void* d_out, int out_size, void* d_ws, size_t ws_size,
                              hipStream_t stream) {
  const float* data_in = (const float*)d_in[0];   // (C=32, N) f32
  const int*   neigh   = (const int*)  d_in[1];   // (N, K=27) int32
  float*       out     = (float*)d_out;           // (C, K, N) f32

  const int N = in_sizes[0] / OCT_C;              // 262144 for this problem

  dim3 block(BLOCK);
  dim3 grid((N + TILE_N - 1) / TILE_N, OCT_K);    // (256, 27) -> ~55K wave32s
  octree2col_gather<<<grid, block, 0, stream>>>(data_in, neigh, out, N);
}